// Cloth_45775761441027
// MI455X (gfx1250) — compile-verified
//
#include <hip/hip_runtime.h>

#define HH 2048
#define WW 2048
#define FRAME_ELEMS (800*800*3)   // 1,920,000 floats
#define NP_ELEMS    (HH*WW*2)     // 8,388,608 floats
#define NBLOCKS     ((HH*WW)/256) // 16384
#define GRAV  9.8f
#define STIFF 4.0f
#define ALPHA 0.003f
#define DECAY 0.99997f

#if defined(__has_builtin)
#if __has_builtin(__builtin_amdgcn_global_load_async_to_lds_b64) && \
    __has_builtin(__builtin_amdgcn_s_wait_asynccnt)
#define USE_ASYNC_LDS 1
#endif
#if __has_builtin(__builtin_amdgcn_wmma_f32_16x16x4_f32)
#define USE_WMMA_REDUCE 1
#endif
#endif

// compile-time path markers (visible in hipcc stderr)
#ifdef USE_ASYNC_LDS
#warning "CDNA5 path: global_load_async_to_lds_b64 ACTIVE"
#else
#warning "CDNA5 path: async LDS builtin MISSING - sync fallback in use"
#endif
#ifdef USE_WMMA_REDUCE
#warning "CDNA5 path: wmma_f32_16x16x4_f32 reduction ACTIVE"
#else
#warning "CDNA5 path: wmma builtin MISSING - shfl fallback in use"
#endif

typedef int v2i __attribute__((ext_vector_type(2)));

#ifdef USE_ASYNC_LDS
typedef __attribute__((address_space(1))) v2i glb_v2i;   // printed as "int2 __device__ *"
typedef __attribute__((address_space(3))) v2i lds_v2i;   // printed as "int2 __shared__ *"
#define AS_GLB(p) ((glb_v2i*)(p))
#define AS_LDS(p) ((lds_v2i*)(p))
#endif

typedef __attribute__((ext_vector_type(2))) float v2f;
typedef __attribute__((ext_vector_type(8))) float v8f;

// ---------------- frame clear ----------------
__global__ __launch_bounds__(256) void k_zero_frame(float4* __restrict__ f4, int n4) {
    int i = blockIdx.x * 256 + threadIdx.x;
    if (i < n4) f4[i] = make_float4(0.f, 0.f, 0.f, 0.f);
}

// ---------------- pass 1: stencil -> diff + per-block energy partial ----------------
__global__ __launch_bounds__(256) void k_physics(const float* __restrict__ pos,
                                                 const float* __restrict__ prev,
                                                 float* __restrict__ diff_out,   // newpos region of d_out
                                                 float* __restrict__ partials) {
    __shared__ float2 s_mid[258];   // row i,   cols jbase-1 .. jbase+256
    __shared__ float2 s_up[256];    // row i-1, cols jbase .. jbase+255
    __shared__ float2 s_dn[256];    // row i+1
    __shared__ float  red[256];

    const int t = threadIdx.x;
    const int i = blockIdx.x >> 3;
    const int jbase = (blockIdx.x & 7) << 8;
    const int j = jbase + t;
    const int cidx = i * WW + j;

    const int iu = (i > 0)      ? i - 1 : i;   // clamped; invalid rows masked below
    const int id = (i < HH - 1) ? i + 1 : i;

    const float2* pos2 = (const float2*)pos;

#ifdef USE_ASYNC_LDS
    // CDNA5 async global->LDS staging (ASYNCcnt path)
    __builtin_amdgcn_global_load_async_to_lds_b64(AS_GLB(&pos2[iu * WW + j]), AS_LDS(&s_up[t]),      0, 0);
    __builtin_amdgcn_global_load_async_to_lds_b64(AS_GLB(&pos2[id * WW + j]), AS_LDS(&s_dn[t]),      0, 0);
    __builtin_amdgcn_global_load_async_to_lds_b64(AS_GLB(&pos2[cidx]),        AS_LDS(&s_mid[t + 1]), 0, 0);
    if (t == 0) {
        int jl = (j > 0) ? j - 1 : j;
        __builtin_amdgcn_global_load_async_to_lds_b64(AS_GLB(&pos2[i * WW + jl]), AS_LDS(&s_mid[0]), 0, 0);
    }
    if (t == 255) {
        int jr = (j < WW - 1) ? j + 1 : j;
        __builtin_amdgcn_global_load_async_to_lds_b64(AS_GLB(&pos2[i * WW + jr]), AS_LDS(&s_mid[257]), 0, 0);
    }
    __builtin_amdgcn_s_wait_asynccnt(0);
    __syncthreads();
#else
    s_up[t]      = pos2[iu * WW + j];
    s_dn[t]      = pos2[id * WW + j];
    s_mid[t + 1] = pos2[cidx];
    if (t == 0)   { int jl = (j > 0)      ? j - 1 : j; s_mid[0]   = pos2[i * WW + jl]; }
    if (t == 255) { int jr = (j < WW - 1) ? j + 1 : j; s_mid[257] = pos2[i * WW + jr]; }
    __syncthreads();
#endif

    const float2 p  = s_mid[t + 1];
    const float2 pp = ((const float2*)prev)[cidx];

    float fx = -GRAV, fy = 0.f;                       // gravity on component 0
    if (j > 0)      { float2 q = s_mid[t];     fx += STIFF * (q.x - p.x); fy += STIFF * (q.y - p.y); }
    if (j < WW - 1) { float2 q = s_mid[t + 2]; fx += STIFF * (q.x - p.x); fy += STIFF * (q.y - p.y); }
    if (i > 0)      { float2 q = s_up[t];      fx += STIFF * (q.x - p.x); fy += STIFF * (q.y - p.y); }
    if (i < HH - 1) { float2 q = s_dn[t];      fx += STIFF * (q.x - p.x); fy += STIFF * (q.y - p.y); }

    // pinned nodes: row 0 every 9th col; row H-1 every 9th col in first half
    bool pin = ((i == 0) && (j % 9 == 0)) ||
               ((i == HH - 1) && (j < WW / 2) && (j % 9 == 0));
    if (pin) { fx = 0.f; fy = 0.f; }

    const float dx = p.x - pp.x + fx;                 // DT=1, MASS=1
    const float dy = p.y - pp.y + fy;

    float2 d; d.x = dx; d.y = dy;
    ((float2*)diff_out)[cidx] = d;

    // deterministic per-block energy partial
    red[t] = dx * dx + dy * dy;
    __syncthreads();
    for (int s = 128; s > 0; s >>= 1) {
        if (t < s) red[t] += red[t + s];
        __syncthreads();
    }
    if (t == 0) partials[blockIdx.x] = red[0];
}

// ---------------- pass 2: deterministic reduction -> scale + new_energy_l ----------------
__global__ __launch_bounds__(256) void k_reduce(const float* __restrict__ partials,
                                                const float* __restrict__ energy_l_in,
                                                float* __restrict__ scale_out,
                                                float* __restrict__ new_el_out) {
    __shared__ float wsum[8];
    const int t = threadIdx.x;

    float p = 0.f;
    for (int k = t; k < NBLOCKS; k += 256) p += partials[k];   // 64 fixed-order adds

    float s;
#ifdef USE_WMMA_REDUCE
    // 32-lane collapse on the matrix pipe: A(16x4) carries lane partials in K0/K2
    // slots, B = ones => D rows broadcast row-sums; lanes<16 hold sum(rows 0..7),
    // lanes>=16 hold sum(rows 8..15); one xor-16 shuffle completes the wave sum.
    v2f a; a.x = p;   a.y = 0.f;
    v2f b; b.x = 1.f; b.y = 1.f;
    v8f c = {};
    c = __builtin_amdgcn_wmma_f32_16x16x4_f32(false, a, false, b, (short)0, c, false, false);
    s = c[0] + c[1] + c[2] + c[3] + c[4] + c[5] + c[6] + c[7];
    s += __shfl_xor(s, 16, 32);
#else
    s = p;
    for (int off = 16; off > 0; off >>= 1) s += __shfl_xor(s, off, 32);
#endif

    if ((t & 31) == 0) wsum[t >> 5] = s;
    __syncthreads();
    if (t == 0) {
        float energy = 0.f;
        for (int k = 0; k < 8; ++k) energy += wsum[k];
        const float el = *energy_l_in;
        float en = fminf(energy, el) * DECAY;
        en = en * 0.8f + energy * 0.2f;
        *scale_out  = en / (energy + 1e-6f);
        *new_el_out = el * (1.f - ALPHA) + en * ALPHA;
    }
}

// ---------------- pass 3: apply scale, write newpos, scatter render ----------------
__global__ __launch_bounds__(256) void k_finalize(const float* __restrict__ pos,
                                                  float* __restrict__ np_io,     // in: diff, out: newpos
                                                  const float* __restrict__ scale_p,
                                                  float* __restrict__ frame) {
    const int idx = blockIdx.x * 256 + threadIdx.x;
    const float2* pos2 = (const float2*)pos;
    float2* np2 = (float2*)np_io;

    const float2 d = np2[idx];
    const float2 p = pos2[idx];
    const float scale = *scale_p;

    const float vn  = sqrtf(d.x * d.x + d.y * d.y);
    const float vel = vn * scale;
    const float k   = vel / fmaxf(vn, 1e-12f);        // vel_dir * vel, matching ref

    float2 np; np.x = p.x + d.x * k; np.y = p.y + d.y * k;
    np2[idx] = np;

    // render: x from component 1, y from component 0
    float xf = np.y / (float)WW * 784.f + 10.f;       // (IMG-20)=784
    float yf = np.x / (float)HH * 104.f + 690.f;      // (IMG-700)=104
    xf = fminf(fmaxf(xf, 0.f), 803.f);
    yf = fminf(fmaxf(yf, 0.f), 803.f);
    const int xi = (int)xf;
    const int yi = (int)yf;
    if (xi >= 2 && xi < 802 && yi >= 2 && yi < 802) {
        // all writers store the same value -> race is benign & deterministic
        frame[((yi - 2) * 800 + (xi - 2)) * 3 + 1] = 255.f;   // green channel
    }
}

extern "C" void kernel_launch(void* const* d_in, const int* in_sizes, int n_in,
                              void* d_out, int out_size, void* d_ws, size_t ws_size,
                              hipStream_t stream) {
    const float* pos  = (const float*)d_in[0];
    const float* prev = (const float*)d_in[1];
    const float* el   = (const float*)d_in[2];

    float* out    = (float*)d_out;
    float* frame  = out;                       // 800*800*3
    float* np     = out + FRAME_ELEMS;         // 2048*2048*2 (diff, then newpos)
    float* new_el = np + NP_ELEMS;             // 1

    float* partials = (float*)d_ws;            // 16384 floats
    float* scale    = partials + NBLOCKS;      // 1 float

    k_zero_frame<<<(FRAME_ELEMS / 4 + 255) / 256, 256, 0, stream>>>((float4*)frame, FRAME_ELEMS / 4);
    k_physics  <<<NBLOCKS, 256, 0, stream>>>(pos, prev, np, partials);
    k_reduce   <<<1,       256, 0, stream>>>(partials, el, scale, new_el);
    k_finalize <<<NBLOCKS, 256, 0, stream>>>(pos, np, scale, frame);
}